// ApproximateLinear_52106543235770
// MI455X (gfx1250) — compile-verified
//
#include <hip/hip_runtime.h>
#include <stdint.h>

// Problem constants (from reference)
#define B_DIM   1024
#define D_DIM   1024
#define N_DB    65536
#define M_SUB   8
#define KS      256
#define DS_SUB  128
#define MV      8
#define DV      256
#define OUT_DIM (MV*DV)   // 2048
#define K_TOP   64

typedef __attribute__((ext_vector_type(2))) float v2f;
typedef __attribute__((ext_vector_type(8))) float v8f;
typedef __attribute__((ext_vector_type(4))) unsigned int v4u;
typedef __attribute__((ext_vector_type(4))) int v4i;
typedef __attribute__((ext_vector_type(8))) int v8i;

#ifndef __has_builtin
#define __has_builtin(x) 0
#endif

#if __has_builtin(__builtin_amdgcn_wmma_f32_16x16x4_f32)
#define HAVE_WMMA_F32X4 1
#else
#define HAVE_WMMA_F32X4 0
#endif

#if __has_builtin(__builtin_amdgcn_tensor_load_to_lds)
#define HAVE_TDM 1
#else
#define HAVE_TDM 0
#endif

// Definitive verification, device compilation only (__HIP_DEVICE_COMPILE__ is
// set solely for the offload pass; __AMDGCN__ proved unreliable here — it is
// visible in this harness's host job too). If either CDNA5 path is missing on
// the device pass, fail loudly instead of silently compiling a fallback.
#if defined(__HIP_DEVICE_COMPILE__) && !HAVE_WMMA_F32X4
#error "gfx1250 device pass lacks __builtin_amdgcn_wmma_f32_16x16x4_f32"
#endif
#if defined(__HIP_DEVICE_COMPILE__) && !HAVE_TDM
#error "gfx1250 device pass lacks __builtin_amdgcn_tensor_load_to_lds"
#endif

// therock-10.0 headers ship only with the amdgpu-toolchain (6-arg TDM builtin);
// ROCm 7.2 clang-22 uses the 5-arg form.
#if defined(__has_include)
# if __has_include(<hip/amd_detail/amd_gfx1250_TDM.h>)
#  define TDM_6ARG 1
# endif
#endif
#ifndef TDM_6ARG
#define TDM_6ARG 0
#endif

__device__ __forceinline__ v8f wmma_f32x4(v2f a, v2f b, v8f c) {
#if HAVE_WMMA_F32X4
  // 8 args: (neg_a, A, neg_b, B, c_mod, C, reuse_a, reuse_b)
  return __builtin_amdgcn_wmma_f32_16x16x4_f32(false, a, false, b, (short)0, c, false, false);
#else
  c[0] += a[0] * b[0];   // host-pass placeholder
  return c;
#endif
}

__device__ __forceinline__ void tdm_wait() {
#if __has_builtin(__builtin_amdgcn_s_wait_tensorcnt)
  __builtin_amdgcn_s_wait_tensorcnt(0);
#endif
}

// ---------------------------------------------------------------------------
// 2D f32 tile load via the Tensor Data Mover.
// D# packing per CDNA5 ISA:
//   group0: [1:0]=count=1 | [63:32]=lds byte addr | [120:64]=global addr | [127:126]=type 2
//   group1: [17:16]=data_size(2 => 4B) | [20]=pad_enable | [24:22]=pad_interval code
//           [31:25]=pad_amount code(0 => 1 dword) | dims/tile/stride bitfields below.
// tensor dims == tile dims (tile fully in-bounds, no OOB clipping).
// pad_code: -1 = none; 5 = every 64 dwords (+1) -> row stride 65;
//           6 = every 128 dwords (+1) -> row stride 129 (LDS bank-conflict kill).
// ---------------------------------------------------------------------------
__device__ __forceinline__ void tdm_load_tile_f32(
    float* lds_ptr, const float* gptr,
    unsigned tile_w, unsigned tile_h,
    unsigned gstride_elems, unsigned lds_stride_elems, int pad_code) {
#if HAVE_TDM
  const unsigned long long ga = (unsigned long long)(uintptr_t)gptr;
  // generic LDS pointer: low 32 bits are the workgroup LDS byte offset
  const unsigned lds_off = (unsigned)(uintptr_t)lds_ptr;

  v4u g0;
  g0[0] = 1u;                                    // count=1, user descriptor
  g0[1] = lds_off;                               // lds_addr
  g0[2] = (unsigned)ga;                          // global_addr[31:0]
  g0[3] = (unsigned)(ga >> 32) | (2u << 30);     // global_addr[56:32] | type=2

  const unsigned long long s0 = gstride_elems;   // tensor_dim0_stride (elements)
  const unsigned pad_en = (pad_code >= 0) ? 1u : 0u;
  const unsigned pic    = (pad_code >= 0) ? (unsigned)pad_code : 0u;

  v8i g1;
  g1[0] = (int)((2u << 16) | (pad_en << 20) | (pic << 22));        // mask=0,dsize=4B,pad
  g1[1] = (int)((tile_w & 0xFFFFu) << 16);                         // tensor_dim0[15:0]
  g1[2] = (int)((tile_w >> 16) | ((tile_h & 0xFFFFu) << 16));      // dim0[31:16]|dim1[15:0]
  g1[3] = (int)((tile_h >> 16) | (tile_w << 16));                  // dim1[31:16]|tile_dim0
  g1[4] = (int)(tile_h & 0xFFFFu);                                 // tile_dim1, tile_dim2=0
  g1[5] = (int)(unsigned)s0;                                       // stride0[31:0]
  g1[6] = (int)((unsigned)(s0 >> 32) & 0xFFFFu);                   // stride0[47:32]
  g1[7] = 0;

  v4i z4 = {0, 0, 0, 0};
#if TDM_6ARG
  v8i z8 = {0, 0, 0, 0, 0, 0, 0, 0};
  __builtin_amdgcn_tensor_load_to_lds(g0, g1, z4, z4, z8, 0);
#else
  __builtin_amdgcn_tensor_load_to_lds(g0, g1, z4, z4, 0);
#endif
#else
  // fallback (host pass only): copy with the calling wave's 32 lanes
  for (unsigned i = (threadIdx.x & 31); i < tile_w * tile_h; i += 32) {
    unsigned r = i / tile_w, c = i - r * tile_w;
    lds_ptr[r * lds_stride_elems + c] = gptr[(size_t)r * gstride_elems + c];
  }
#endif
  (void)lds_stride_elems;
}

// ---------------------------------------------------------------------------
// K0: q = x @ W. Block = 4 waves -> 16x64 tile. K chunked by 64, TDM double-
// buffered into LDS (X tile padded to stride 65), WMMA f32 16x16x4 from LDS.
// ---------------------------------------------------------------------------
#define WT_BK 64
#define XS_STR 65

__global__ __launch_bounds__(128)
void al_whiten_wmma(const float* __restrict__ X, const float* __restrict__ W,
                    float* __restrict__ Q) {
  __shared__ float xs[2][16 * XS_STR];   // 16 x 64 (+1 pad per row)
  __shared__ float ws[2][WT_BK * 64];    // 64 x 64
  const int wave = threadIdx.x >> 5;
  const int lane = threadIdx.x & 31;
  const int lr = lane & 15, lh = lane >> 4;
  const int row0 = blockIdx.x * 16;
  const int colb = blockIdx.y * 64;
  const int NT = D_DIM / WT_BK;          // 16 K-slabs

  if (wave == 0) {
    tdm_load_tile_f32(xs[0], X + (size_t)row0 * D_DIM, WT_BK, 16, D_DIM, XS_STR, 5);
    tdm_load_tile_f32(ws[0], W + colb, 64, WT_BK, D_DIM, 64, -1);
    tdm_wait();
  }
  __syncthreads();

  v8f acc0 = {}, acc1 = {};
  for (int kt = 0; kt < NT; ++kt) {
    const int cur = kt & 1, nxt = cur ^ 1;
    if (wave == 0 && kt + 1 < NT) {      // prefetch next slab while computing
      tdm_load_tile_f32(xs[nxt], X + (size_t)row0 * D_DIM + (kt + 1) * WT_BK,
                        WT_BK, 16, D_DIM, XS_STR, 5);
      tdm_load_tile_f32(ws[nxt], W + (size_t)(kt + 1) * WT_BK * D_DIM + colb,
                        64, WT_BK, D_DIM, 64, -1);
    }
    const float* xr = &xs[cur][lr * XS_STR];
    const float* wt = &ws[cur][wave * 16 + lr];
#pragma unroll
    for (int k = 0; k < WT_BK; k += 8) {
      v2f a, b;
      int kk = k + 2 * lh;
      a.x = xr[kk];      a.y = xr[kk + 1];
      b.x = wt[kk * 64]; b.y = wt[(kk + 1) * 64];
      acc0 = wmma_f32x4(a, b, acc0);
      kk = k + 4 + 2 * lh;
      a.x = xr[kk];      a.y = xr[kk + 1];
      b.x = wt[kk * 64]; b.y = wt[(kk + 1) * 64];
      acc1 = wmma_f32x4(a, b, acc1);
    }
    if (kt + 1 < NT) {
      if (wave == 0) tdm_wait();
      __syncthreads();                   // nxt buffer visible / cur reusable
    }
  }
  v8f acc = acc0 + acc1;
#pragma unroll
  for (int r = 0; r < 8; ++r)
    Q[(size_t)(row0 + r + 8 * lh) * D_DIM + colb + wave * 16 + lr] = acc[r];
}

// ---------------------------------------------------------------------------
// K1: squared norms
// ---------------------------------------------------------------------------
__global__ __launch_bounds__(256)
void al_norms(const float* __restrict__ Q, const float* __restrict__ KCB,
              float* __restrict__ qsq, float* __restrict__ csq) {
  int gid = blockIdx.x * blockDim.x + threadIdx.x;
  const int NQ = B_DIM * M_SUB;
  const int NC = M_SUB * KS;
  if (gid < NQ) {
    const float* p = Q + (size_t)gid * DS_SUB;
    float s = 0.f;
#pragma unroll 8
    for (int i = 0; i < DS_SUB; ++i) { float v = p[i]; s += v * v; }
    qsq[gid] = s;
  } else if (gid < NQ + NC) {
    int j = gid - NQ;
    const float* p = KCB + (size_t)j * DS_SUB;
    float s = 0.f;
#pragma unroll 8
    for (int i = 0; i < DS_SUB; ++i) { float v = p[i]; s += v * v; }
    csq[j] = s;
  }
}

// ---------------------------------------------------------------------------
// K2: LUT[b][m][c] = qsq + csq - 2*cross. Single-shot TDM stage: q tile
// (16x128, pad->129) and codebook tile (64x128, pad->129, also fixes the
// transpose-read bank pattern), then 32 WMMA K-steps entirely from LDS.
// ---------------------------------------------------------------------------
#define QT_STR 129

__global__ __launch_bounds__(128)
void al_lut_wmma(const float* __restrict__ Q, const float* __restrict__ KCB,
                 const float* __restrict__ qsq, const float* __restrict__ csq,
                 float* __restrict__ LUT) {
  __shared__ float qt[16 * QT_STR];
  __shared__ float ct[64 * QT_STR];
  const int wave = threadIdx.x >> 5;
  const int lane = threadIdx.x & 31;
  const int lr = lane & 15, lh = lane >> 4;
  const int m  = blockIdx.z;
  const int b0 = blockIdx.x * 16;
  const int cb = blockIdx.y * 64;

  if (wave == 0) {
    tdm_load_tile_f32(qt, Q + (size_t)b0 * D_DIM + m * DS_SUB,
                      DS_SUB, 16, D_DIM, QT_STR, 6);
    tdm_load_tile_f32(ct, KCB + ((size_t)m * KS + cb) * DS_SUB,
                      DS_SUB, 64, DS_SUB, QT_STR, 6);
    tdm_wait();
  }
  __syncthreads();

  const float* qr = &qt[lr * QT_STR];
  const float* cr = &ct[(wave * 16 + lr) * QT_STR];
  v8f acc0 = {}, acc1 = {};
#pragma unroll
  for (int k = 0; k < DS_SUB; k += 8) {
    v2f a, b;
    int kk = k + 2 * lh;
    a.x = qr[kk]; a.y = qr[kk + 1];
    b.x = cr[kk]; b.y = cr[kk + 1];     // B[k][c] = codebook[m][c][k]
    acc0 = wmma_f32x4(a, b, acc0);
    kk = k + 4 + 2 * lh;
    a.x = qr[kk]; a.y = qr[kk + 1];
    b.x = cr[kk]; b.y = cr[kk + 1];
    acc1 = wmma_f32x4(a, b, acc1);
  }
  v8f acc = acc0 + acc1;
#pragma unroll
  for (int r = 0; r < 8; ++r) {
    int b = b0 + r + 8 * lh;
    int c = cb + wave * 16 + lr;
    LUT[((size_t)b * M_SUB + m) * KS + c] =
        qsq[b * M_SUB + m] + csq[m * KS + c] - 2.0f * acc[r];
  }
}

// ---------------------------------------------------------------------------
// K3: ADC scan + top-64 select, one workgroup per b-row. Exploits 320KB
// LDS/WGP: 8KB LUT + full 65536-entry key array + 8 per-wave histograms
// resident (278.6KB). 2-level radix select + collect. Per-wave histogram
// privatization removes 8-way inter-wave LDS atomic contention in pass 1.
// ---------------------------------------------------------------------------
#define SCAN_THREADS 256
#define SCAN_WAVES   (SCAN_THREADS / 32)
#define SCAN_SMEM_BYTES (M_SUB*KS*4 + N_DB*4 + SCAN_WAVES*256*4 + 64)

__global__ __launch_bounds__(SCAN_THREADS)
void al_scan_topk(const float* __restrict__ LUT, const int* __restrict__ key_codes,
                  float* __restrict__ topk_dist, int* __restrict__ topk_idx) {
  extern __shared__ unsigned char smem[];
  float*    lut_s = (float*)smem;
  unsigned* keys  = (unsigned*)(smem + M_SUB * KS * 4);
  unsigned* hist  = (unsigned*)(smem + M_SUB * KS * 4 + N_DB * 4); // 8x256 slices
  unsigned* ctl   = hist + SCAN_WAVES * 256;

  const int b = blockIdx.x;
  const int tid = threadIdx.x;
  const int wave = tid >> 5;

  const float* lb = LUT + (size_t)b * (M_SUB * KS);
  for (int i = tid; i < M_SUB * KS; i += SCAN_THREADS) lut_s[i] = lb[i];
  for (int i = tid; i < SCAN_WAVES * 256; i += SCAN_THREADS) hist[i] = 0;
  if (tid < 8) ctl[tid] = 0;
  __syncthreads();

  // Pass 1: fused gather-sum distances -> monotonic keys in LDS,
  // per-wave byte-0 histograms
  unsigned* myhist = hist + wave * 256;
  for (int n = tid; n < N_DB; n += SCAN_THREADS) {
    const int4* crow = (const int4*)(key_codes + (size_t)n * M_SUB);
    __builtin_prefetch(key_codes + (size_t)(n + 4 * SCAN_THREADS) * M_SUB, 0, 0);
    int4 cA = crow[0], cB = crow[1];
    float d = lut_s[cA.x]          + lut_s[KS     + cA.y]
            + lut_s[2 * KS + cA.z] + lut_s[3 * KS + cA.w]
            + lut_s[4 * KS + cB.x] + lut_s[5 * KS + cB.y]
            + lut_s[6 * KS + cB.z] + lut_s[7 * KS + cB.w];
    unsigned u = __float_as_uint(d);
    unsigned key = u ^ ((unsigned)((int)u >> 31) | 0x80000000u);
    keys[n] = key;
    atomicAdd(&myhist[key >> 24], 1u);
  }
  __syncthreads();

  // merge per-wave slices into slice 0
  for (int i = tid; i < 256; i += SCAN_THREADS) {
    unsigned s = 0;
#pragma unroll
    for (int w = 0; w < SCAN_WAVES; ++w) s += hist[w * 256 + i];
    hist[i] = s;
  }
  __syncthreads();

  if (tid == 0) {  // byte-0 bin of the 64th smallest
    unsigned cum = 0, s1 = 255, c1 = 0;
    for (int i = 0; i < 256; ++i) {
      if (cum + hist[i] >= (unsigned)K_TOP) { s1 = (unsigned)i; c1 = cum; break; }
      cum += hist[i];
    }
    ctl[0] = s1; ctl[1] = c1;
  }
  __syncthreads();
  const unsigned s1 = ctl[0], C1 = ctl[1];
  for (int i = tid; i < 256; i += SCAN_THREADS) hist[i] = 0;
  __syncthreads();

  // Pass 2: byte-1 histogram within boundary bin (few elements -> shared hist)
  for (int n = tid; n < N_DB; n += SCAN_THREADS) {
    unsigned key = keys[n];
    if ((key >> 24) == s1) atomicAdd(&hist[(key >> 16) & 255u], 1u);
  }
  __syncthreads();

  if (tid == 0) {
    unsigned need = (unsigned)K_TOP - C1;
    unsigned cum = 0, s2 = 255, clt = C1;
    for (int i = 0; i < 256; ++i) {
      if (cum + hist[i] >= need) { s2 = (unsigned)i; clt = C1 + cum; break; }
      cum += hist[i];
    }
    ctl[2] = (s1 << 8) | s2;   // 16-bit boundary prefix
    ctl[3] = clt;              // exact count strictly below boundary
    ctl[4] = 0; ctl[5] = 0;    // collect counters
  }
  __syncthreads();
  const unsigned bnd = ctl[2], Clt = ctl[3];

  float* td = topk_dist + (size_t)b * K_TOP;
  int*   ti = topk_idx  + (size_t)b * K_TOP;

  // Pass 3: collect exactly 64 winners (boundary ties first-come)
  for (int n = tid; n < N_DB; n += SCAN_THREADS) {
    unsigned key = keys[n];
    unsigned p16 = key >> 16;
    int pos = -1;
    if (p16 < bnd)       pos = (int)atomicAdd(&ctl[4], 1u);
    else if (p16 == bnd) pos = (int)(Clt + atomicAdd(&ctl[5], 1u));
    if (pos >= 0 && pos < K_TOP) {
      unsigned u = (key & 0x80000000u) ? (key ^ 0x80000000u) : ~key;
      td[pos] = __uint_as_float(u);
      ti[pos] = n;
    }
  }
}

// ---------------------------------------------------------------------------
// K4: softmax over -dist + weighted value-codebook gather
// ---------------------------------------------------------------------------
__global__ __launch_bounds__(256)
void al_output(const float* __restrict__ topk_dist, const int* __restrict__ topk_idx,
               const int* __restrict__ value_codes, const float* __restrict__ VCB,
               const float* __restrict__ bias, float* __restrict__ Y) {
  __shared__ float w_s[K_TOP];
  __shared__ int   vc_s[K_TOP][MV];
  const int b = blockIdx.x, tid = threadIdx.x;

  if (tid < K_TOP) w_s[tid] = -topk_dist[(size_t)b * K_TOP + tid];
  for (int i = tid; i < K_TOP * MV; i += 256) {
    int k = i / MV, m = i % MV;
    vc_s[k][m] = value_codes[(size_t)topk_idx[(size_t)b * K_TOP + k] * MV + m];
  }
  __syncthreads();
  if (tid == 0) {
    float mx = w_s[0];
    for (int i = 1; i < K_TOP; ++i) mx = fmaxf(mx, w_s[i]);
    float sum = 0.f;
    for (int i = 0; i < K_TOP; ++i) { float e = __expf(w_s[i] - mx); w_s[i] = e; sum += e; }
    float inv = 1.0f / sum;
    for (int i = 0; i < K_TOP; ++i) w_s[i] *= inv;
  }
  __syncthreads();

  float acc[MV];
#pragma unroll
  for (int j = 0; j < MV; ++j) acc[j] = 0.f;
  for (int k = 0; k < K_TOP; ++k) {
    float w = w_s[k];
#pragma unroll
    for (int j = 0; j < MV; ++j)
      acc[j] += w * VCB[((size_t)j * KS + vc_s[k][j]) * DV + tid];
  }
#pragma unroll
  for (int j = 0; j < MV; ++j) {
    int o = j * DV + tid;
    Y[(size_t)b * OUT_DIM + o] = acc[j] + bias[o];
  }
}

// ---------------------------------------------------------------------------
extern "C" void kernel_launch(void* const* d_in, const int* in_sizes, int n_in,
                              void* d_out, int out_size, void* d_ws, size_t ws_size,
                              hipStream_t stream) {
  const float* x    = (const float*)d_in[0];
  const float* W    = (const float*)d_in[1];
  const float* KCB  = (const float*)d_in[2];
  const float* VCB  = (const float*)d_in[3];
  const float* bias = (const float*)d_in[4];
  const int*   kcodes = (const int*)d_in[5];
  const int*   vcodes = (const int*)d_in[6];
  float* Y = (float*)d_out;

  float* q   = (float*)d_ws;
  float* lut = q   + (size_t)B_DIM * D_DIM;
  float* qsq = lut + (size_t)B_DIM * M_SUB * KS;
  float* csq = qsq + (size_t)B_DIM * M_SUB;
  float* tkd = csq + (size_t)M_SUB * KS;
  int*   tki = (int*)(tkd + (size_t)B_DIM * K_TOP);

  dim3 g0(B_DIM / 16, D_DIM / 64);
  al_whiten_wmma<<<g0, 128, 0, stream>>>(x, W, q);

  int n_norm = B_DIM * M_SUB + M_SUB * KS;
  al_norms<<<(n_norm + 255) / 256, 256, 0, stream>>>(q, KCB, qsq, csq);

  dim3 g2(B_DIM / 16, KS / 64, M_SUB);
  al_lut_wmma<<<g2, 128, 0, stream>>>(q, KCB, qsq, csq, lut);

  (void)hipFuncSetAttribute(reinterpret_cast<const void*>(al_scan_topk),
                            hipFuncAttributeMaxDynamicSharedMemorySize,
                            SCAN_SMEM_BYTES);
  al_scan_topk<<<B_DIM, SCAN_THREADS, SCAN_SMEM_BYTES, stream>>>(lut, kcodes, tkd, tki);

  al_output<<<B_DIM, 256, 0, stream>>>(tkd, tki, vcodes, VCB, bias, Y);
}